// GCRNGRU_33285996544264
// MI455X (gfx1250) — compile-verified
//
#include <hip/hip_runtime.h>
#include <hip/hip_bf16.h>
#include <math.h>

// MI455X / gfx1250, wave32. FP32 WMMA (V_WMMA_F32_16X16X4_F32) for the dense
// GEMMs + GLOBAL_LOAD_ASYNC_TO_LDS_B128 for operand staging (ASYNCcnt path).

typedef __attribute__((ext_vector_type(2))) float v2f;
typedef __attribute__((ext_vector_type(8))) float v8f;
typedef int v4i_async __attribute__((ext_vector_type(4)));
typedef __attribute__((address_space(1))) v4i_async* gvec_p;   // global 16B vector ptr
typedef __attribute__((address_space(3))) v4i_async* lvec_p;   // LDS 16B vector ptr

#define HDIM 128
#define LDSW 132   // padded LDS row stride (mod 64 = 4 -> conflict-free 16x4 A-frag b64 reads)

#if defined(__has_builtin)
#if __has_builtin(__builtin_amdgcn_global_load_async_to_lds_b128)
#define HAVE_ASYNC_LDS 1
#endif
#endif

__device__ __forceinline__ float sigmoidf_(float x) {
    return 1.0f / (1.0f + __expf(-x));
}

#ifdef HAVE_ASYNC_LDS
__device__ __forceinline__ void async_cp16(float* lds, const float* g) {
    // GLOBAL_LOAD_ASYNC_TO_LDS_B128: per-lane 16B, global -> LDS, no VGPR round-trip
    __builtin_amdgcn_global_load_async_to_lds_b128((gvec_p)(g), (lvec_p)(lds), 0, 0);
}
#endif

__device__ __forceinline__ void wait_async0() {
#if defined(__has_builtin) && __has_builtin(__builtin_amdgcn_s_wait_asynccnt)
    __builtin_amdgcn_s_wait_asynccnt(0);
#else
    asm volatile("s_wait_asynccnt 0x0" ::: "memory");
#endif
}

// ---------------------------------------------------------------- zero fill
__global__ void k_zero(float* __restrict__ p, size_t n) {
    size_t i = (size_t)blockIdx.x * blockDim.x + threadIdx.x;
    size_t stride = (size_t)gridDim.x * blockDim.x;
    for (; i < n; i += stride) p[i] = 0.0f;
}

// ---------------------------------------------------------------- degree
__global__ void k_deg(const int* __restrict__ ei, float* __restrict__ deg, int E) {
    int e = blockIdx.x * blockDim.x + threadIdx.x;
    if (e < E) atomicAdd(&deg[ei[e]], 1.0f);   // src row of edge_index
}

// ---------------------------------------------------------------- dinv
__global__ void k_dinv(const float* __restrict__ deg, float* __restrict__ dinv, int N) {
    int i = blockIdx.x * blockDim.x + threadIdx.x;
    if (i < N) {
        float d = deg[i];
        dinv[i] = (d > 0.0f) ? __frsqrt_rn(fmaxf(d, 1.0f)) : 0.0f;
    }
}

// ---------------------------------------------------------------- Xh = x @ pre_w + pre_b
// 8 waves/block; each wave: 16 rows x all 8 column tiles. A staged in LDS.
__global__ __launch_bounds__(256) void k_pre_gemm(
    const float* __restrict__ x, const float* __restrict__ pre_w,
    const float* __restrict__ pre_b, float* __restrict__ Xh, int N)
{
    __shared__ float sA[128 * LDSW];
    const int tid  = threadIdx.x;
    const int lane = tid & 31, wave = tid >> 5;
    const int rowBase = blockIdx.x * 128;

    // stage 128x128 block of x into LDS (padded stride 132)
#pragma unroll
    for (int i = 0; i < 16; ++i) {
        int lin = tid + i * 256;          // float4 index, 4096 total
        int r   = lin >> 5;               // 32 float4 per row
        int c4  = (lin & 31) << 2;
        float*       lp = &sA[r * LDSW + c4];
        const float* gp = x + (size_t)(rowBase + r) * HDIM + c4;
#ifdef HAVE_ASYNC_LDS
        if (rowBase + r < N) async_cp16(lp, gp);
        else                 *(float4*)lp = make_float4(0.f, 0.f, 0.f, 0.f);
#else
        float4 v = make_float4(0.f, 0.f, 0.f, 0.f);
        if (rowBase + r < N) v = *(const float4*)gp;
        *(float4*)lp = v;
#endif
    }
#ifdef HAVE_ASYNC_LDS
    wait_async0();
#endif
    __syncthreads();

    const int mr = wave * 16 + (lane & 15);   // A-frag row in LDS
    const int kh = (lane >> 4) << 1;          // K phase: 0 (lanes 0-15) / 2 (lanes 16-31)
    const int nn = lane & 15;
    const int half = (lane >> 4) * 8;

    for (int n0 = 0; n0 < HDIM; n0 += 16) {
        v8f acc = {};
#pragma unroll
        for (int kk = 0; kk < HDIM; kk += 4) {
            v2f a, b;
            const float* ap = &sA[mr * LDSW + kk + kh];
            a.x = ap[0]; a.y = ap[1];                       // ds_load_b64
            const float* bp = pre_w + (size_t)(kk + kh) * HDIM + n0 + nn;
            b.x = bp[0]; b.y = bp[HDIM];
            acc = __builtin_amdgcn_wmma_f32_16x16x4_f32(
                false, a, false, b, (short)0, acc, false, false);
        }
        float bias = pre_b[n0 + nn];
#pragma unroll
        for (int i = 0; i < 8; ++i) {
            int row = rowBase + wave * 16 + half + i;
            if (row < N) Xh[(size_t)row * HDIM + n0 + nn] = acc[i] + bias;
        }
    }
}

// ---------------------------------------------------------------- LX += w * Xh[src] scattered at dst
// one wave per edge; lane handles a float4 of the 128-wide feature row.
__global__ __launch_bounds__(256) void k_spmm(
    const float* __restrict__ Xh, const int* __restrict__ ei,
    const float* __restrict__ dinv, float* __restrict__ LX, int E)
{
    int lane = threadIdx.x & 31;
    int e = (int)(((size_t)blockIdx.x * 256 + threadIdx.x) >> 5);
    if (e >= E) return;
    int src = ei[e];
    int dst = ei[E + e];
    float w = -dinv[src] * dinv[dst];
    const float4 v = *(const float4*)(Xh + (size_t)src * HDIM + lane * 4);
    float* o = LX + (size_t)dst * HDIM + lane * 4;
    atomicAdd(o + 0, w * v.x);
    atomicAdd(o + 1, w * v.y);
    atomicAdd(o + 2, w * v.z);
    atomicAdd(o + 3, w * v.w);
}

// ---------------------------------------------------------------- fused gates
// Z  = sigmoid(Xh@Wx[0][0] + LX@Wx[0][1] + bx[0] + bh[0])
// Ht = tanh   (Xh@Wx[2][0] + LX@Wx[2][1] + bx[2] + bh[2])
// h  = relu((1-Z)*Ht)        (Hc == 0 folds all Wh GEMMs away)
__global__ __launch_bounds__(256) void k_gates(
    const float* __restrict__ Xh, const float* __restrict__ LX,
    const float* __restrict__ Wx, const float* __restrict__ bx,
    const float* __restrict__ bh, float* __restrict__ hOut, int N)
{
    __shared__ float sX[128 * LDSW];
    __shared__ float sL[128 * LDSW];
    const int tid  = threadIdx.x;
    const int lane = tid & 31, wave = tid >> 5;
    const int rowBase = blockIdx.x * 128;

#pragma unroll
    for (int i = 0; i < 16; ++i) {
        int lin = tid + i * 256;
        int r   = lin >> 5;
        int c4  = (lin & 31) << 2;
        float*       lpx = &sX[r * LDSW + c4];
        float*       lpl = &sL[r * LDSW + c4];
        const float* gpx = Xh + (size_t)(rowBase + r) * HDIM + c4;
        const float* gpl = LX + (size_t)(rowBase + r) * HDIM + c4;
#ifdef HAVE_ASYNC_LDS
        if (rowBase + r < N) {
            async_cp16(lpx, gpx);
            async_cp16(lpl, gpl);
        } else {
            *(float4*)lpx = make_float4(0.f, 0.f, 0.f, 0.f);
            *(float4*)lpl = make_float4(0.f, 0.f, 0.f, 0.f);
        }
#else
        float4 vx = make_float4(0.f, 0.f, 0.f, 0.f);
        float4 vl = make_float4(0.f, 0.f, 0.f, 0.f);
        if (rowBase + r < N) {
            vx = *(const float4*)gpx;
            vl = *(const float4*)gpl;
        }
        *(float4*)lpx = vx;
        *(float4*)lpl = vl;
#endif
    }
#ifdef HAVE_ASYNC_LDS
    wait_async0();
#endif
    __syncthreads();

    const float* Wz0 = Wx + (size_t)(0 * 2 + 0) * HDIM * HDIM;
    const float* Wz1 = Wx + (size_t)(0 * 2 + 1) * HDIM * HDIM;
    const float* Wt0 = Wx + (size_t)(2 * 2 + 0) * HDIM * HDIM;
    const float* Wt1 = Wx + (size_t)(2 * 2 + 1) * HDIM * HDIM;

    const int mr = wave * 16 + (lane & 15);
    const int kh = (lane >> 4) << 1;
    const int nn = lane & 15;
    const int half = (lane >> 4) * 8;

    for (int n0 = 0; n0 < HDIM; n0 += 16) {
        v8f accZ = {};
        v8f accT = {};
#pragma unroll
        for (int kk = 0; kk < HDIM; kk += 4) {
            v2f aX, aL, bz0, bz1, bt0, bt1;
            const float* axp = &sX[mr * LDSW + kk + kh];
            const float* alp = &sL[mr * LDSW + kk + kh];
            aX.x = axp[0]; aX.y = axp[1];
            aL.x = alp[0]; aL.y = alp[1];
            size_t boff = (size_t)(kk + kh) * HDIM + n0 + nn;
            bz0.x = Wz0[boff]; bz0.y = Wz0[boff + HDIM];
            bz1.x = Wz1[boff]; bz1.y = Wz1[boff + HDIM];
            bt0.x = Wt0[boff]; bt0.y = Wt0[boff + HDIM];
            bt1.x = Wt1[boff]; bt1.y = Wt1[boff + HDIM];
            accZ = __builtin_amdgcn_wmma_f32_16x16x4_f32(false, aX, false, bz0, (short)0, accZ, false, false);
            accT = __builtin_amdgcn_wmma_f32_16x16x4_f32(false, aX, false, bt0, (short)0, accT, false, false);
            accZ = __builtin_amdgcn_wmma_f32_16x16x4_f32(false, aL, false, bz1, (short)0, accZ, false, false);
            accT = __builtin_amdgcn_wmma_f32_16x16x4_f32(false, aL, false, bt1, (short)0, accT, false, false);
        }
        float bz = bx[0 * HDIM + n0 + nn] + bh[0 * HDIM + n0 + nn];
        float bt = bx[2 * HDIM + n0 + nn] + bh[2 * HDIM + n0 + nn];
#pragma unroll
        for (int i = 0; i < 8; ++i) {
            int row = rowBase + wave * 16 + half + i;
            if (row < N) {
                float z  = sigmoidf_(accZ[i] + bz);
                float ht = tanhf(accT[i] + bt);
                float hv = (1.0f - z) * ht;
                hOut[(size_t)row * HDIM + n0 + nn] = fmaxf(hv, 0.0f);
            }
        }
    }
}

// ---------------------------------------------------------------- link decoder
// out[e] = sum_k h[a,k]*h[b,k]*(pw[k,0]+pw[k,1]) + (pb0+pb1)
__global__ __launch_bounds__(256) void k_decode(
    const float* __restrict__ h, const int* __restrict__ eli,
    const float* __restrict__ post_w, const float* __restrict__ post_b,
    float* __restrict__ out, int EL)
{
    int lane = threadIdx.x & 31;
    int e = (int)(((size_t)blockIdx.x * 256 + threadIdx.x) >> 5);
    if (e >= EL) return;
    int a = eli[e];
    int b = eli[EL + e];
    int f = lane * 4;
    float4 ha = *(const float4*)(h + (size_t)a * HDIM + f);
    float4 hb = *(const float4*)(h + (size_t)b * HDIM + f);
    float s = ha.x * hb.x * (post_w[(f + 0) * 2] + post_w[(f + 0) * 2 + 1])
            + ha.y * hb.y * (post_w[(f + 1) * 2] + post_w[(f + 1) * 2 + 1])
            + ha.z * hb.z * (post_w[(f + 2) * 2] + post_w[(f + 2) * 2 + 1])
            + ha.w * hb.w * (post_w[(f + 3) * 2] + post_w[(f + 3) * 2 + 1]);
#pragma unroll
    for (int off = 16; off >= 1; off >>= 1) s += __shfl_xor(s, off, 32);
    if (lane == 0) out[e] = s + post_b[0] + post_b[1];
}

// ---------------------------------------------------------------- launch
extern "C" void kernel_launch(void* const* d_in, const int* in_sizes, int n_in,
                              void* d_out, int out_size, void* d_ws, size_t ws_size,
                              hipStream_t stream) {
    const float* x      = (const float*)d_in[0];
    const int*   ei     = (const int*)d_in[1];
    const int*   eli    = (const int*)d_in[2];
    const float* pre_w  = (const float*)d_in[3];
    const float* pre_b  = (const float*)d_in[4];
    const float* Wx     = (const float*)d_in[5];
    const float* bx     = (const float*)d_in[6];
    // d_in[7] = Wh: GEMMs fold away since H0 == 0 (only bh biases survive)
    const float* bh     = (const float*)d_in[8];
    const float* post_w = (const float*)d_in[9];
    const float* post_b = (const float*)d_in[10];

    const int N  = in_sizes[0] / HDIM;
    const int E  = in_sizes[1] / 2;
    const int EL = in_sizes[2] / 2;

    // workspace carve-out (256B aligned)
    char* ws = (char*)d_ws;
    size_t off = 0;
    auto carve = [&](size_t bytes) -> float* {
        float* p = (float*)(ws + off);
        off = (off + bytes + 255) & ~(size_t)255;
        return p;
    };
    float* deg  = carve((size_t)N * sizeof(float));
    float* dinv = carve((size_t)N * sizeof(float));
    float* Xh   = carve((size_t)N * HDIM * sizeof(float));
    float* LX   = carve((size_t)N * HDIM * sizeof(float));
    float* hbuf = carve((size_t)N * HDIM * sizeof(float));
    (void)ws_size; (void)n_in; (void)out_size;

    float* out = (float*)d_out;

    k_zero<<<1024, 256, 0, stream>>>(deg, (size_t)N);
    k_zero<<<2048, 256, 0, stream>>>(LX, (size_t)N * HDIM);
    k_deg<<<(E + 255) / 256, 256, 0, stream>>>(ei, deg, E);
    k_dinv<<<(N + 255) / 256, 256, 0, stream>>>(deg, dinv, N);
    k_pre_gemm<<<(N + 127) / 128, 256, 0, stream>>>(x, pre_w, pre_b, Xh, N);
    k_spmm<<<(E + 7) / 8, 256, 0, stream>>>(Xh, ei, dinv, LX, E);
    k_gates<<<(N + 127) / 128, 256, 0, stream>>>(Xh, LX, Wx, bx, bh, hbuf, N);
    k_decode<<<(EL + 7) / 8, 256, 0, stream>>>(hbuf, eli, post_w, post_b, out, EL);
}